// RidgeRegression_55508157333808
// MI455X (gfx1250) — compile-verified
//
#include <hip/hip_runtime.h>
#include <math.h>

// Ridge regression, B=32, T=2048, P=512, D=64, fp32 throughout.
// W_b = (X_b^T X_b + lmb I)^{-1} X_b^T Y_b ; out_b = X_b W_b

typedef __attribute__((ext_vector_type(2))) float v2f;
typedef __attribute__((ext_vector_type(4))) float v4f;
typedef __attribute__((ext_vector_type(8))) float v8f;

#define B_ 32
#define T_ 2048
#define P_ 512
#define D_ 64
#define KC 32

// ---------------------------------------------------------------------------
// Kernel 1: C = X^T [X | Y] per batch.
// grid (P/128, (P+D)/64, B), block 256 (8 waves).
// Workgroup computes a 128x64 tile; each wave a 32x32 strip (2x2 WMMA accs),
// so every A/B fragment is reused across 2 WMMAs.
// ---------------------------------------------------------------------------
__global__ __launch_bounds__(256) void gram_xty_kernel(
    const float* __restrict__ X, const float* __restrict__ Y,
    float* __restrict__ G, float* __restrict__ XtY)
{
  __shared__ __align__(16) float As[KC][128];
  __shared__ __align__(16) float Bs[KC][64];

  const int b     = blockIdx.z;
  const int pBase = blockIdx.x * 128;
  const int qTile = blockIdx.y;          // 0..7 -> Gram cols, 8 -> XtY
  const bool isY  = (qTile == 8);
  const int qBase = isY ? 0 : qTile * 64;
  const float* Xb   = X + (size_t)b * T_ * P_;
  const float* Bsrc = isY ? (Y + (size_t)b * T_ * D_) : Xb;
  const int ldB     = isY ? D_ : P_;

  const int tid   = threadIdx.x;
  const int lane  = tid & 31;
  const int w     = tid >> 5;            // 0..7
  const int mPair = w & 3;               // M tile pair 0..3  (covers 32 rows)
  const int nHalf = w >> 2;              // N tile pair 0..1  (covers 32 cols)
  const int m     = lane & 15;
  const int kHi   = (lane < 16) ? 0 : 2; // fp32 16x4 fragment K offset

  v8f acc00 = {}, acc01 = {}, acc10 = {}, acc11 = {};

  // A-tile fill: 32x128 floats = 1024 float4, 4 per thread
  const int aCol = (tid & 31) * 4;       // 0..124
  const int aRow = tid >> 5;             // 0..7
  // B-tile fill: 32x64 floats = 512 float4, 2 per thread
  const int bCol = (tid & 15) * 4;       // 0..60
  const int bRow = tid >> 4;             // 0..15

  for (int k0 = 0; k0 < T_; k0 += KC) {
#pragma unroll
    for (int i = 0; i < 4; ++i) {
      const int k = aRow + 8 * i;
      *(v4f*)&As[k][aCol] = *(const v4f*)&Xb[(size_t)(k0 + k) * P_ + pBase + aCol];
    }
#pragma unroll
    for (int i = 0; i < 2; ++i) {
      const int k = bRow + 16 * i;
      *(v4f*)&Bs[k][bCol] = *(const v4f*)&Bsrc[(size_t)(k0 + k) * ldB + qBase + bCol];
    }
    if (k0 + KC < T_) {  // prefetch next A chunk -> global_prefetch_b8
      __builtin_prefetch(&Xb[(size_t)(k0 + KC + aRow) * P_ + pBase + aCol], 0, 1);
    }
    __syncthreads();
#pragma unroll
    for (int kk = 0; kk < KC; kk += 4) {
      v2f a0, a1, b0, b1;
      a0.x = As[kk + kHi][mPair * 32 + m];
      a0.y = As[kk + kHi + 1][mPair * 32 + m];
      a1.x = As[kk + kHi][mPair * 32 + 16 + m];
      a1.y = As[kk + kHi + 1][mPair * 32 + 16 + m];
      b0.x = Bs[kk + kHi][nHalf * 32 + m];
      b0.y = Bs[kk + kHi + 1][nHalf * 32 + m];
      b1.x = Bs[kk + kHi][nHalf * 32 + 16 + m];
      b1.y = Bs[kk + kHi + 1][nHalf * 32 + 16 + m];
      acc00 = __builtin_amdgcn_wmma_f32_16x16x4_f32(false, a0, false, b0,
                                                    (short)0, acc00, false, false);
      acc01 = __builtin_amdgcn_wmma_f32_16x16x4_f32(false, a0, false, b1,
                                                    (short)0, acc01, false, false);
      acc10 = __builtin_amdgcn_wmma_f32_16x16x4_f32(false, a1, false, b0,
                                                    (short)0, acc10, false, false);
      acc11 = __builtin_amdgcn_wmma_f32_16x16x4_f32(false, a1, false, b1,
                                                    (short)0, acc11, false, false);
    }
    __syncthreads();
  }

  // D layout: VGPR r -> M = r (+8 for lanes 16..31), N = lane%16
  const int hi8 = (lane < 16) ? 0 : 8;
  const int n   = lane & 15;
#pragma unroll
  for (int r = 0; r < 8; ++r) {
    const int row0 = pBase + mPair * 32 + r + hi8;
    const int row1 = row0 + 16;
    const int c0   = nHalf * 32 + n;
    const int c1   = c0 + 16;
    if (!isY) {
      float* Grow0 = G + ((size_t)b * P_ + row0) * P_ + qBase;
      float* Grow1 = G + ((size_t)b * P_ + row1) * P_ + qBase;
      Grow0[c0] = acc00[r];
      Grow0[c1] = acc01[r];
      Grow1[c0] = acc10[r];
      Grow1[c1] = acc11[r];
    } else {
      float* Xr0 = XtY + ((size_t)b * P_ + row0) * D_;
      float* Xr1 = XtY + ((size_t)b * P_ + row1) * D_;
      Xr0[c0] = acc00[r];
      Xr0[c1] = acc01[r];
      Xr1[c0] = acc10[r];
      Xr1[c1] = acc11[r];
    }
  }
}

// ---------------------------------------------------------------------------
// Kernel 2: per-batch Cholesky of (G + lmb I) in-place (lower), then forward
// and backward substitution for the 64 RHS, with Z resident in LDS (128 KB).
// grid (B), block 256, dynamic LDS = (P*D + P + 257) floats.
// ---------------------------------------------------------------------------
__global__ __launch_bounds__(256) void cholesky_solve_kernel(
    float* __restrict__ G, const float* __restrict__ XtY,
    float* __restrict__ W, const float* __restrict__ lmbp)
{
  extern __shared__ float smem[];
  float* Z    = smem;                    // P_*D_ floats (128 KB)
  float* rowj = Z + P_ * D_;             // P_ floats
  float* red  = rowj + P_;               // 257 floats

  const int b   = blockIdx.x;
  const int tid = threadIdx.x;
  const float lmb = lmbp[0];
  float* Gb = G + (size_t)b * P_ * P_;

  // ---- Cholesky: G(+lmb I) = L L^T, L stored in lower triangle ----
  for (int j = 0; j < P_; ++j) {
    float s = 0.f;
    for (int k = tid; k < j; k += 256) {
      const float v = Gb[(size_t)j * P_ + k];
      rowj[k] = v;
      s += v * v;
    }
    red[tid] = s;
    __syncthreads();
    for (int off = 128; off > 0; off >>= 1) {
      if (tid < off) red[tid] += red[tid + off];
      __syncthreads();
    }
    if (tid == 0) {
      const float dval = Gb[(size_t)j * P_ + j] + lmb - red[0];
      const float Ljj  = sqrtf(fmaxf(dval, 1e-20f));
      Gb[(size_t)j * P_ + j] = Ljj;
      red[256] = 1.0f / Ljj;
    }
    __syncthreads();
    const float invL = red[256];
    for (int i = j + 1 + tid; i < P_; i += 256) {
      float acc = Gb[(size_t)i * P_ + j];
      for (int k = 0; k < j; ++k) acc -= Gb[(size_t)i * P_ + k] * rowj[k];
      Gb[(size_t)i * P_ + j] = acc * invL;
    }
    __syncthreads();
  }

  // ---- Load RHS into LDS ----
  const float* XtYb = XtY + (size_t)b * P_ * D_;
  for (int idx = tid; idx < P_ * D_; idx += 256) Z[idx] = XtYb[idx];
  __syncthreads();

  const int d  = tid & 63;   // RHS column
  const int ig = tid >> 6;   // row group 0..3

  // ---- Forward: L Z' = Z ----
  for (int j = 0; j < P_; ++j) {
    const float Ljj = Gb[(size_t)j * P_ + j];
    if (tid < D_) Z[j * D_ + tid] /= Ljj;
    __syncthreads();
    const float zj = Z[j * D_ + d];
    for (int i = j + 1 + ig; i < P_; i += 4)
      Z[i * D_ + d] -= Gb[(size_t)i * P_ + j] * zj;
    __syncthreads();
  }

  // ---- Backward: L^T W = Z' ----
  for (int j = P_ - 1; j >= 0; --j) {
    const float Ljj = Gb[(size_t)j * P_ + j];
    if (tid < D_) Z[j * D_ + tid] /= Ljj;
    __syncthreads();
    const float wj = Z[j * D_ + d];
    for (int i = ig; i < j; i += 4)
      Z[i * D_ + d] -= Gb[(size_t)j * P_ + i] * wj;
    __syncthreads();
  }

  float* Wb = W + (size_t)b * P_ * D_;
  for (int idx = tid; idx < P_ * D_; idx += 256) Wb[idx] = Z[idx];
}

// ---------------------------------------------------------------------------
// Kernel 3: out = X * W per batch. grid (T/128, 1, B), block 256.
// Workgroup computes a 128x64 output tile; each wave 4 accumulators sharing
// one A fragment per WMMA step.
// ---------------------------------------------------------------------------
__global__ __launch_bounds__(256) void xw_kernel(
    const float* __restrict__ X, const float* __restrict__ W,
    float* __restrict__ Out)
{
  __shared__ __align__(16) float Xs[128][KC];   // t-major: Xs[t][k]
  __shared__ __align__(16) float Ws[KC][D_];    // k-major: Ws[k][d]

  const int b  = blockIdx.z;
  const int t0 = blockIdx.x * 128;
  const float* Xb = X + (size_t)b * T_ * P_;
  const float* Wb = W + (size_t)b * P_ * D_;

  const int tid  = threadIdx.x;
  const int lane = tid & 31;
  const int w    = tid >> 5;             // M sub-tile 0..7
  const int m    = lane & 15;
  const int kHi  = (lane < 16) ? 0 : 2;

  v8f acc[4];
  {
    v8f z = {};
#pragma unroll
    for (int j = 0; j < 4; ++j) acc[j] = z;
  }

  // Xs fill: 128x32 = 1024 float4, 4 per thread
  const int xCol = (tid & 7) * 4;        // 0..28
  const int xRow = tid >> 3;             // 0..31
  // Ws fill: 32x64 = 512 float4, 2 per thread
  const int wCol = (tid & 15) * 4;       // 0..60
  const int wRow = tid >> 4;             // 0..15

  for (int k0 = 0; k0 < P_; k0 += KC) {
#pragma unroll
    for (int i = 0; i < 4; ++i) {
      const int t = xRow + 32 * i;       // 0..127
      *(v4f*)&Xs[t][xCol] = *(const v4f*)&Xb[(size_t)(t0 + t) * P_ + k0 + xCol];
    }
#pragma unroll
    for (int i = 0; i < 2; ++i) {
      const int k = wRow + 16 * i;       // 0..31
      *(v4f*)&Ws[k][wCol] = *(const v4f*)&Wb[(size_t)(k0 + k) * D_ + wCol];
    }
    if (k0 + KC < P_) {  // prefetch next X chunk
      __builtin_prefetch(&Xb[(size_t)(t0 + xRow) * P_ + k0 + KC + xCol], 0, 1);
    }
    __syncthreads();
#pragma unroll
    for (int kk = 0; kk < KC; kk += 4) {
      v2f a;
      a.x = Xs[w * 16 + m][kk + kHi];
      a.y = Xs[w * 16 + m][kk + kHi + 1];
#pragma unroll
      for (int j = 0; j < 4; ++j) {
        v2f bb;
        bb.x = Ws[kk + kHi][j * 16 + m];
        bb.y = Ws[kk + kHi + 1][j * 16 + m];
        acc[j] = __builtin_amdgcn_wmma_f32_16x16x4_f32(false, a, false, bb,
                                                       (short)0, acc[j],
                                                       false, false);
      }
    }
    __syncthreads();
  }

  const int hi8 = (lane < 16) ? 0 : 8;
  const int n   = lane & 15;
#pragma unroll
  for (int r = 0; r < 8; ++r) {
    const int t = t0 + w * 16 + r + hi8;
#pragma unroll
    for (int j = 0; j < 4; ++j)
      Out[((size_t)b * T_ + t) * D_ + j * 16 + n] = acc[j][r];
  }
}

// ---------------------------------------------------------------------------
extern "C" void kernel_launch(void* const* d_in, const int* in_sizes, int n_in,
                              void* d_out, int out_size, void* d_ws, size_t ws_size,
                              hipStream_t stream) {
  (void)in_sizes; (void)n_in; (void)out_size; (void)ws_size;
  const float* X   = (const float*)d_in[0];   // [B,T,P]
  const float* Y   = (const float*)d_in[1];   // [B,T,D]
  const float* lmb = (const float*)d_in[2];   // scalar
  float* Out = (float*)d_out;                 // [B,T,D]

  // Workspace layout: G [B,P,P] | XtY [B,P,D] | W [B,P,D]  (~42 MB total)
  float* G   = (float*)d_ws;
  float* XtY = G + (size_t)B_ * P_ * P_;
  float* W   = XtY + (size_t)B_ * P_ * D_;

  dim3 g1(P_ / 128, (P_ + D_) / 64, B_);      // (4, 9, 32)
  gram_xty_kernel<<<g1, 256, 0, stream>>>(X, Y, G, XtY);

  const size_t ldsBytes = (size_t)(P_ * D_ + P_ + 257) * sizeof(float);
  cholesky_solve_kernel<<<dim3(B_), 256, ldsBytes, stream>>>(G, XtY, W, lmb);

  dim3 g3(T_ / 128, 1, B_);                   // (16, 1, 32)
  xw_kernel<<<g3, 256, 0, stream>>>(X, W, Out);
}